// MGD_SeqOnly_11484742550061
// MI455X (gfx1250) — compile-verified
//
#include <hip/hip_runtime.h>
#include <math.h>

// Problem constants (match reference)
#define NN   20000
#define LL   50
#define FF   32
#define HH   128
#define RHH  64
#define EE   320000
#define NPAD 66        // padded LDS row stride for GRU hidden state (even -> 8B-aligned pair loads)

typedef float v2f __attribute__((ext_vector_type(2)));
typedef float v8f __attribute__((ext_vector_type(8)));

__device__ __forceinline__ v8f vz8() {
    v8f v = {0.f,0.f,0.f,0.f,0.f,0.f,0.f,0.f};
    return v;
}

// D = A(16x4,f32) * B(4x16,f32) + C   -> v_wmma_f32_16x16x4_f32
__device__ __forceinline__ v8f wmma4(v2f a, v2f b, v8f c) {
    return __builtin_amdgcn_wmma_f32_16x16x4_f32(false, a, false, b, (short)0, c, false, false);
}

__device__ __forceinline__ float sigm(float x) { return 1.0f / (1.0f + __expf(-x)); }

// ---------------------------------------------------------------------------
// Utility kernels
// ---------------------------------------------------------------------------
__global__ void zero_kernel(float* __restrict__ p, long n) {
    long i = (long)blockIdx.x * blockDim.x + threadIdx.x;
    long stride = (long)gridDim.x * blockDim.x;
    for (; i < n; i += stride) p[i] = 0.0f;
}

__global__ void degree_kernel(const int* __restrict__ ei,
                              float* __restrict__ degT, float* __restrict__ degS) {
    int e = blockIdx.x * blockDim.x + threadIdx.x;
    if (e >= EE) return;
    atomicAdd(&degT[ei[EE + e]], 1.0f);   // edges landing at tgt
    atomicAdd(&degS[ei[e]], 1.0f);        // edges leaving src
}

// Wd = Wc_bot - Wc_top  (Wc is row-major [256,128])
__global__ void wd_kernel(const float* __restrict__ Wc, float* __restrict__ Wd) {
    int i = blockIdx.x * blockDim.x + threadIdx.x;
    if (i < 128 * 128) Wd[i] = Wc[16384 + i] - Wc[i];
}

// St[tgt] += x[src]; Ss[src] += x[tgt]  (128 floats per edge, chunked by 32)
__global__ void scatter_kernel(const int* __restrict__ ei, const float* __restrict__ x,
                               float* __restrict__ St, float* __restrict__ Ss) {
    long id = (long)blockIdx.x * blockDim.x + threadIdx.x;
    if (id >= (long)EE * 4) return;
    int e    = (int)(id >> 2);
    int part = ((int)id & 3) * 32;
    int s = ei[e], t = ei[EE + e];
    const float* xs = x + (size_t)s * HH + part;
    const float* xt = x + (size_t)t * HH + part;
    __builtin_prefetch(xs, 0, 1);
    __builtin_prefetch(xt, 0, 1);
    float* st = St + (size_t)t * HH + part;
    float* ss = Ss + (size_t)s * HH + part;
#pragma unroll
    for (int q = 0; q < 8; ++q) {
        float4 a = *(const float4*)(xs + 4 * q);
        atomicAdd(st + 4*q + 0, a.x); atomicAdd(st + 4*q + 1, a.y);
        atomicAdd(st + 4*q + 2, a.z); atomicAdd(st + 4*q + 3, a.w);
        float4 b = *(const float4*)(xt + 4 * q);
        atomicAdd(ss + 4*q + 0, b.x); atomicAdd(ss + 4*q + 1, b.y);
        atomicAdd(ss + 4*q + 2, b.z); atomicAdd(ss + 4*q + 3, b.w);
    }
}

// ---------------------------------------------------------------------------
// Generic C[nrows,128] = A[nrows,128] @ W[128,128]  (raw, no bias) via WMMA f32
// Block = 256 threads = 8 waves; each wave owns a 16-row tile.
// W staged in LDS pair-interleaved so B fragments are single 8B loads.
// ---------------------------------------------------------------------------
__global__ void gemm128_kernel(const float* __restrict__ A, const float* __restrict__ W,
                               float* __restrict__ C, int nrows) {
    extern __shared__ float wl[];                    // 128*128 floats, pair-interleaved
    const int tid = threadIdx.x;
    for (int i = tid; i < 128 * 128; i += blockDim.x) {
        int k = i >> 7, c = i & 127;
        wl[((k >> 1) * 128 + c) * 2 + (k & 1)] = W[i];
    }
    __syncthreads();

    const int wave = tid >> 5, lane = tid & 31;
    const int row0 = (blockIdx.x * (blockDim.x >> 5) + wave) * 16;
    if (row0 >= nrows) return;
    const int m16 = lane & 15, hi = lane >> 4, koff = 2 * hi;

    v8f acc[8];
#pragma unroll
    for (int ct = 0; ct < 8; ++ct) acc[ct] = vz8();

    const float* ar = A + (size_t)(row0 + m16) * 128 + koff;
#pragma unroll 4
    for (int ks = 0; ks < 32; ++ks) {
        v2f a = *(const v2f*)(ar + 4 * ks);          // A[m16][4ks+koff .. +1]
        const int kp = 2 * ks + hi;
#pragma unroll
        for (int ct = 0; ct < 8; ++ct) {
            v2f b = *(const v2f*)(wl + (kp * 128 + m16 + 16 * ct) * 2);
            acc[ct] = wmma4(a, b, acc[ct]);
        }
    }
#pragma unroll
    for (int ct = 0; ct < 8; ++ct)
#pragma unroll
        for (int v = 0; v < 8; ++v)
            C[(size_t)(row0 + v + 8 * hi) * 128 + m16 + 16 * ct] = acc[ct][v];
}

// ---------------------------------------------------------------------------
// GRU (packed-sequence semantics) + LayerNorm epilogue, WMMA f32 recurrence.
// Block = 128 threads = 4 waves; each wave scans a 16-node tile for 50 steps.
// h lives in per-wave LDS; weights pair-interleaved in LDS.
// Gate columns {c, c+64, c+128} share a lane/slot -> fused gate math on accums.
// ---------------------------------------------------------------------------
__global__ void gru_kernel(const float* __restrict__ seq, const int* __restrict__ len,
                           const float* __restrict__ Wih, const float* __restrict__ Whh,
                           const float* __restrict__ bih, const float* __restrict__ bhh,
                           const float* __restrict__ lng, const float* __restrict__ lnb,
                           float* __restrict__ xout, int colOff) {
    extern __shared__ float sm[];
    float* whh_l = sm;                   // 32 pairs * 192 * 2 = 12288
    float* wih_l = whh_l + 12288;        // 16 pairs * 192 * 2 = 6144
    float* bih_l = wih_l + 6144;         // 192
    float* bhh_l = bih_l + 192;          // 192
    float* lng_l = bhh_l + 192;          // 64
    float* lnb_l = lng_l + 64;           // 64
    float* hbase = lnb_l + 64;           // 4 waves * 16 * NPAD

    const int tid = threadIdx.x;
    for (int i = tid; i < 64 * 192; i += blockDim.x) {
        int k = i / 192, c = i % 192;
        whh_l[((k >> 1) * 192 + c) * 2 + (k & 1)] = Whh[i];
    }
    for (int i = tid; i < 32 * 192; i += blockDim.x) {
        int k = i / 192, c = i % 192;
        wih_l[((k >> 1) * 192 + c) * 2 + (k & 1)] = Wih[i];
    }
    for (int i = tid; i < 192; i += blockDim.x) { bih_l[i] = bih[i]; bhh_l[i] = bhh[i]; }
    for (int i = tid; i < 64;  i += blockDim.x) { lng_l[i] = lng[i]; lnb_l[i] = lnb[i]; }
    __syncthreads();

    const int wave = tid >> 5, lane = tid & 31;
    const int tile = blockIdx.x * (blockDim.x >> 5) + wave;
    if (tile >= NN / 16) return;
    const int row0 = tile * 16;
    const int m16 = lane & 15, hi = lane >> 4, koff = 2 * hi;
    float* hbuf = hbase + wave * 16 * NPAD;

    for (int i = lane; i < 16 * RHH; i += 32) hbuf[(i >> 6) * NPAD + (i & 63)] = 0.0f;

    int lenr[8];
#pragma unroll
    for (int v = 0; v < 8; ++v) lenr[v] = len[row0 + v + 8 * hi];

    float br_i[4], bz_i[4], bn_i[4], br_h[4], bz_h[4], bn_h[4];
#pragma unroll
    for (int ct = 0; ct < 4; ++ct) {
        int c = m16 + 16 * ct;
        br_i[ct] = bih_l[c]; bz_i[ct] = bih_l[c + 64]; bn_i[ct] = bih_l[c + 128];
        br_h[ct] = bhh_l[c]; bz_h[ct] = bhh_l[c + 64]; bn_h[ct] = bhh_l[c + 128];
    }

    const float* gxbase = seq + (size_t)(row0 + m16) * LL * FF + koff;

    for (int t = 0; t < LL; ++t) {
        v2f xa[8], ha[16];
        const float* gx = gxbase + (size_t)t * FF;
#pragma unroll
        for (int ks = 0; ks < 8; ++ks)  xa[ks] = *(const v2f*)(gx + 4 * ks);
        const float* hrow = hbuf + m16 * NPAD + koff;
#pragma unroll
        for (int ks = 0; ks < 16; ++ks) ha[ks] = *(const v2f*)(hrow + 4 * ks);

        v8f hn[4];
#pragma unroll
        for (int tt = 0; tt < 4; ++tt) {
            const int cr = m16 + 16 * tt, cz = cr + 64, cn = cr + 128;
            v8f xr = vz8(), xz = vz8(), xn = vz8();
            v8f gr = vz8(), gz = vz8(), gn = vz8();
#pragma unroll
            for (int ks = 0; ks < 8; ++ks) {                 // XP = x_t @ Wih (K=32)
                const int kp = 2 * ks + hi;
                xr = wmma4(xa[ks], *(const v2f*)(wih_l + (kp * 192 + cr) * 2), xr);
                xz = wmma4(xa[ks], *(const v2f*)(wih_l + (kp * 192 + cz) * 2), xz);
                xn = wmma4(xa[ks], *(const v2f*)(wih_l + (kp * 192 + cn) * 2), xn);
            }
#pragma unroll
            for (int ks = 0; ks < 16; ++ks) {                // G = h @ Whh (K=64)
                const int kp = 2 * ks + hi;
                gr = wmma4(ha[ks], *(const v2f*)(whh_l + (kp * 192 + cr) * 2), gr);
                gz = wmma4(ha[ks], *(const v2f*)(whh_l + (kp * 192 + cz) * 2), gz);
                gn = wmma4(ha[ks], *(const v2f*)(whh_l + (kp * 192 + cn) * 2), gn);
            }
#pragma unroll
            for (int v = 0; v < 8; ++v) {                    // fused gates
                const int m = v + 8 * hi;
                const float hold = hbuf[m * NPAD + cr];
                const float r  = sigm(xr[v] + br_i[tt] + gr[v] + br_h[tt]);
                const float z  = sigm(xz[v] + bz_i[tt] + gz[v] + bz_h[tt]);
                const float nu = tanhf(xn[v] + bn_i[tt] + r * (gn[v] + bn_h[tt]));
                hn[tt][v] = (t < lenr[v]) ? ((1.0f - z) * nu + z * hold) : hold;
            }
        }
#pragma unroll
        for (int tt = 0; tt < 4; ++tt)
#pragma unroll
            for (int v = 0; v < 8; ++v)
                hbuf[(v + 8 * hi) * NPAD + m16 + 16 * tt] = hn[tt][v];
    }

    // LayerNorm over 64 dims + write into x[:, colOff:colOff+64]
    if (hi == 0) {
        const float* hr = hbuf + m16 * NPAD;
        float s = 0.0f;
        for (int c = 0; c < 64; ++c) s += hr[c];
        const float mu = s * (1.0f / 64.0f);
        float vv = 0.0f;
        for (int c = 0; c < 64; ++c) { float d = hr[c] - mu; vv += d * d; }
        const float rs = rsqrtf(vv * (1.0f / 64.0f) + 1e-5f);
        float* xo = xout + (size_t)(row0 + m16) * HH + colOff;
        for (int c = 0; c < 64; ++c) xo[c] = (hr[c] - mu) * rs * lng_l[c] + lnb_l[c];
    }
}

// ---------------------------------------------------------------------------
// CataConv epilogue: build 3 views, attention weights, weighted sum, LN, ReLU.
// mv0 = SV+bs; mv1 = degT*P + QT + bc; mv2 = degS*P + QS + bc
// Streaming two-pass: pass1 gets A1-dots + gram matrix + sums, pass2 emits.
// ---------------------------------------------------------------------------
__global__ void attn_kernel(const float* __restrict__ SV, const float* __restrict__ P,
                            const float* __restrict__ QT, const float* __restrict__ QS,
                            const float* __restrict__ degT, const float* __restrict__ degS,
                            const float* __restrict__ bs, const float* __restrict__ bc,
                            const float* __restrict__ A1, const float* __restrict__ b1,
                            const float* __restrict__ A2,
                            const float* __restrict__ lng, const float* __restrict__ lnb,
                            float* __restrict__ xout, float* __restrict__ attOut) {
    __shared__ float A1l[128 * 16];
    __shared__ float A2l[16], b1l[16], bsl[128], bcl[128], lngl[128], lnbl[128];
    const int tid = threadIdx.x;
    for (int i = tid; i < 2048; i += blockDim.x) A1l[i] = A1[i];
    if (tid < 16)  { A2l[tid] = A2[tid]; b1l[tid] = b1[tid]; }
    if (tid < 128) { bsl[tid] = bs[tid]; bcl[tid] = bc[tid]; lngl[tid] = lng[tid]; lnbl[tid] = lnb[tid]; }
    __syncthreads();

    const int n = blockIdx.x * blockDim.x + tid;
    if (n >= NN) return;
    const float dt = degT[n], dsg = degS[n];
    const float* sv = SV + (size_t)n * 128;
    const float* pp = P  + (size_t)n * 128;
    const float* qt = QT + (size_t)n * 128;
    const float* qs = QS + (size_t)n * 128;

    float u0[16], u1[16], u2[16];
#pragma unroll
    for (int j = 0; j < 16; ++j) { u0[j] = b1l[j]; u1[j] = b1l[j]; u2[j] = b1l[j]; }
    float s0 = 0, s1 = 0, s2 = 0;
    float g00 = 0, g01 = 0, g02 = 0, g11 = 0, g12 = 0, g22 = 0;
    for (int d = 0; d < 128; ++d) {
        const float m0 = sv[d] + bsl[d];
        const float pv = pp[d];
        const float m1 = dt  * pv + qt[d] + bcl[d];
        const float m2 = dsg * pv + qs[d] + bcl[d];
        s0 += m0; s1 += m1; s2 += m2;
        g00 += m0 * m0; g01 += m0 * m1; g02 += m0 * m2;
        g11 += m1 * m1; g12 += m1 * m2; g22 += m2 * m2;
#pragma unroll
        for (int j = 0; j < 16; ++j) {
            const float a = A1l[d * 16 + j];
            u0[j] += m0 * a; u1[j] += m1 * a; u2[j] += m2 * a;
        }
    }
    float t0 = 0, t1 = 0, t2 = 0;
#pragma unroll
    for (int j = 0; j < 16; ++j) {
        t0 += tanhf(u0[j]) * A2l[j];
        t1 += tanhf(u1[j]) * A2l[j];
        t2 += tanhf(u2[j]) * A2l[j];
    }
    const float w0 = tanhf(t0), w1 = tanhf(t1), w2 = tanhf(t2);
    const float mean = (w0 * s0 + w1 * s1 + w2 * s2) * (1.0f / 128.0f);
    const float ex2 = (w0*w0*g00 + w1*w1*g11 + w2*w2*g22 +
                       2.0f * (w0*w1*g01 + w0*w2*g02 + w1*w2*g12)) * (1.0f / 128.0f);
    const float rs = rsqrtf(fmaxf(ex2 - mean * mean, 0.0f) + 1e-5f);

    float* xo = xout + (size_t)n * 128;
    for (int d = 0; d < 128; ++d) {
        const float m0 = sv[d] + bsl[d];
        const float pv = pp[d];
        const float m1 = dt  * pv + qt[d] + bcl[d];
        const float m2 = dsg * pv + qs[d] + bcl[d];
        const float xn = w0 * m0 + w1 * m1 + w2 * m2;
        const float y  = (xn - mean) * rs * lngl[d] + lnbl[d];
        xo[d] = fmaxf(y, 0.0f);
    }
    if (attOut) { attOut[n * 3 + 0] = w0; attOut[n * 3 + 1] = w1; attOut[n * 3 + 2] = w2; }
}

// ---------------------------------------------------------------------------
// Decoder tail: h1 = relu(H1raw + b1); z = h1 @ W2 + b2; log_softmax(z)
// ---------------------------------------------------------------------------
__global__ void dec2_kernel(const float* __restrict__ H1, const float* __restrict__ db1,
                            const float* __restrict__ W2, const float* __restrict__ db2,
                            float* __restrict__ out) {
    __shared__ float w2l[256], b1l[128], b2l[2];
    const int tid = threadIdx.x;
    if (tid < 256) w2l[tid] = W2[tid];
    if (tid < 128) b1l[tid] = db1[tid];
    if (tid < 2)   b2l[tid] = db2[tid];
    __syncthreads();
    const int n = blockIdx.x * blockDim.x + tid;
    if (n >= NN) return;
    const float* h = H1 + (size_t)n * 128;
    float z0 = b2l[0], z1 = b2l[1];
    for (int d = 0; d < 128; ++d) {
        const float v = fmaxf(h[d] + b1l[d], 0.0f);
        z0 += v * w2l[2 * d];
        z1 += v * w2l[2 * d + 1];
    }
    const float m = fmaxf(z0, z1);
    const float lse = m + logf(__expf(z0 - m) + __expf(z1 - m));
    out[2 * n + 0] = z0 - lse;
    out[2 * n + 1] = z1 - lse;
}

// ---------------------------------------------------------------------------
// Host launch
// ---------------------------------------------------------------------------
extern "C" void kernel_launch(void* const* d_in, const int* in_sizes, int n_in,
                              void* d_out, int out_size, void* d_ws, size_t ws_size,
                              hipStream_t stream) {
    (void)in_sizes; (void)n_in; (void)out_size; (void)ws_size;

    const float* in_seq   = (const float*)d_in[0];
    const float* out_seq  = (const float*)d_in[1];
    const int*   len_in   = (const int*)d_in[2];
    const int*   len_out  = (const int*)d_in[3];
    const int*   ei       = (const int*)d_in[4];
    const float* gi_Wih   = (const float*)d_in[5];
    const float* gi_Whh   = (const float*)d_in[6];
    const float* gi_bih   = (const float*)d_in[7];
    const float* gi_bhh   = (const float*)d_in[8];
    const float* go_Wih   = (const float*)d_in[9];
    const float* go_Whh   = (const float*)d_in[10];
    const float* go_bih   = (const float*)d_in[11];
    const float* go_bhh   = (const float*)d_in[12];
    const float* ln_in_g  = (const float*)d_in[13];
    const float* ln_in_b  = (const float*)d_in[14];
    const float* ln_out_g = (const float*)d_in[15];
    const float* ln_out_b = (const float*)d_in[16];
    const float* dec1_W   = (const float*)d_in[35];
    const float* dec1_b   = (const float*)d_in[36];
    const float* dec2_W   = (const float*)d_in[37];
    const float* dec2_b   = (const float*)d_in[38];

    float* ws = (float*)d_ws;
    const size_t NH = (size_t)NN * HH;
    float* X    = ws;            // current node features [N,128]
    float* ST   = ws + 1 * NH;   // scatter sum at tgt
    float* SS   = ws + 2 * NH;   // scatter sum at src
    float* SVb  = ws + 3 * NH;   // x @ Ws
    float* Pb   = ws + 4 * NH;   // x @ Wc_top   (also reused as decoder hidden)
    float* QT   = ws + 5 * NH;   // ST @ Wd
    float* QS   = ws + 6 * NH;   // SS @ Wd
    float* DEGT = ws + 7 * NH;   // [N]
    float* DEGS = DEGT + NN;     // [N]
    float* WD   = DEGS + NN;     // [128,128]
    float* OUT  = (float*)d_out; // logits [N,2] then first_att [N,3]

    const size_t gemm_lds = (size_t)128 * 128 * sizeof(float);
    const size_t gru_lds  = (size_t)(12288 + 6144 + 192 + 192 + 64 + 64 + 4 * 16 * NPAD) * sizeof(float);

    // Degrees (graph is fixed per launch)
    zero_kernel<<<256, 256, 0, stream>>>(DEGT, (long)(2 * NN));
    degree_kernel<<<(EE + 255) / 256, 256, 0, stream>>>(ei, DEGT, DEGS);

    // GRU encoders -> X (cols 0..63 and 64..127)
    const int gruBlocks = (NN / 16 + 3) / 4;  // 4 waves/block, 16 nodes/wave
    gru_kernel<<<gruBlocks, 128, gru_lds, stream>>>(in_seq, len_in, gi_Wih, gi_Whh,
                                                    gi_bih, gi_bhh, ln_in_g, ln_in_b, X, 0);
    gru_kernel<<<gruBlocks, 128, gru_lds, stream>>>(out_seq, len_out, go_Wih, go_Whh,
                                                    go_bih, go_bhh, ln_out_g, ln_out_b, X, RHH);

    const int gemmBlocks = (NN + 127) / 128;  // 8 waves/block, 16 rows/wave
    for (int li = 0; li < 2; ++li) {
        const int b = 17 + 9 * li;
        const float* Ws = (const float*)d_in[b + 0];
        const float* bs = (const float*)d_in[b + 1];
        const float* Wc = (const float*)d_in[b + 2];
        const float* bc = (const float*)d_in[b + 3];
        const float* A1 = (const float*)d_in[b + 4];
        const float* b1 = (const float*)d_in[b + 5];
        const float* A2 = (const float*)d_in[b + 6];
        const float* lg = (const float*)d_in[b + 7];
        const float* lb = (const float*)d_in[b + 8];

        wd_kernel<<<64, 256, 0, stream>>>(Wc, WD);
        zero_kernel<<<512, 256, 0, stream>>>(ST, (long)(2 * NH));        // ST and SS contiguous
        scatter_kernel<<<(EE * 4 + 255) / 256, 256, 0, stream>>>(ei, X, ST, SS);

        gemm128_kernel<<<gemmBlocks, 256, gemm_lds, stream>>>(X,  Ws, SVb, NN);
        gemm128_kernel<<<gemmBlocks, 256, gemm_lds, stream>>>(X,  Wc, Pb,  NN); // uses top 128 rows of Wc
        gemm128_kernel<<<gemmBlocks, 256, gemm_lds, stream>>>(ST, WD, QT,  NN);
        gemm128_kernel<<<gemmBlocks, 256, gemm_lds, stream>>>(SS, WD, QS,  NN);

        attn_kernel<<<(NN + 255) / 256, 256, 0, stream>>>(SVb, Pb, QT, QS, DEGT, DEGS,
                                                          bs, bc, A1, b1, A2, lg, lb,
                                                          X, li == 0 ? OUT + 2 * NN : nullptr);
    }

    // Decoder
    gemm128_kernel<<<gemmBlocks, 256, gemm_lds, stream>>>(X, dec1_W, Pb, NN);
    dec2_kernel<<<(NN + 255) / 256, 256, 0, stream>>>(Pb, dec1_b, dec2_W, dec2_b, OUT);
}